// SparseCorrespondenceMatcher_43284680409103
// MI455X (gfx1250) — compile-verified
//
#include <hip/hip_runtime.h>

// CDNA5 / gfx1250 — wave32, WMMA 16x16x4 F32.
typedef __attribute__((ext_vector_type(2))) float v2f;
typedef __attribute__((ext_vector_type(8))) float v8f;

#define TILE 16
#define TJ_PER_WAVE 16  // j-tiles per wave: reuse A fragment, contiguous row stores

// ---------------------------------------------------------------------------
// Init: keep = 1.0, counts = 0
// ---------------------------------------------------------------------------
__global__ void scm_init_kernel(float* __restrict__ keep, int* __restrict__ counts, int n) {
    int i = blockIdx.x * blockDim.x + threadIdx.x;
    if (i < n) { keep[i] = 1.0f; counts[i] = 0; }
}

// ---------------------------------------------------------------------------
// dist0 tile build via V_WMMA_F32_16X16X4_F32 + fused neighbor counts.
//   A row i = [x_i, y_i, |p_i|^2, 1]
//   B col j = [-2x_j, -2y_j, 1, |p_j|^2]
//   A.B = squared L2 distance; one WMMA per 16x16 tile, no epilogue math.
// Block = (32, 8): 8 waves; each wave owns one i-tile strip x 16 j-tiles.
// Counts are accumulated branchlessly in registers across the whole strip
// and flushed with 8 atomics from lanes 0 and 16 at the end (no EXEC churn
// in the hot loop).
// ---------------------------------------------------------------------------
__global__ __launch_bounds__(256) void scm_dist_wmma_kernel(
    const float* __restrict__ pts,     // [n,2]
    const int*   __restrict__ thr_p,   // scalar threshold (squared)
    float*       __restrict__ dist,    // [n,n] output (dist0)
    int*         __restrict__ counts,  // [n] neighbor counts (atomic accum)
    int n, int ntiles)
{
    const float thr  = (float)(*thr_p);
    const float maxv = thr + 1.0f;

    const int lane = threadIdx.x;       // 0..31 (wave32)
    const int col  = lane & 15;         // column within tile (and point-in-tile idx)
    const int hi   = lane >> 4;         // 0: lanes 0-15 (K=0,1 / rows 0-7), 1: lanes 16-31

    const int ti      = blockIdx.y;
    const int tj_base = (blockIdx.x * blockDim.y + threadIdx.y) * TJ_PER_WAVE;

    // A fragment (16x4 f32 = 2 VGPRs): lanes 0-15 hold K=0,1; lanes 16-31 hold K=2,3.
    const int gi = ti * TILE + col;
    const v2f pi = *(const v2f*)(pts + 2 * (size_t)gi);
    const float ni = fmaf(pi.x, pi.x, pi.y * pi.y);
    v2f a;
    if (hi == 0) { a.x = pi.x; a.y = pi.y; }     // K=0: x_i, K=1: y_i
    else         { a.x = ni;   a.y = 1.0f; }     // K=2: |p_i|^2, K=3: 1

    // Which C VGPR holds this lane's diagonal element (row==col), if any.
    const int vdiag = col - 8 * hi;              // valid when in [0,8)

    // Per-row count accumulators. Meaningful on lane 0 (rows 0..7) and
    // lane 16 (rows 8..15); harmlessly computed by every lane.
    int cnt[8];
    #pragma unroll
    for (int v = 0; v < 8; ++v) cnt[v] = 0;

    for (int t = 0; t < TJ_PER_WAVE; ++t) {
        const int tj = tj_base + t;
        if (tj >= ntiles) break;                 // uniform across the wave

        // B fragment (4x16 f32 = 2 VGPRs), mirrored layout (lane = column N).
        const int gj = tj * TILE + col;
        const v2f pj = *(const v2f*)(pts + 2 * (size_t)gj);
        const float nj = fmaf(pj.x, pj.x, pj.y * pj.y);
        v2f b;
        if (hi == 0) { b.x = -2.0f * pj.x; b.y = -2.0f * pj.y; }  // K=0,1
        else         { b.x = 1.0f;         b.y = nj; }            // K=2,3

        v8f c = {};
        // (neg_a, A, neg_b, B, c_mod, C, reuse_a, reuse_b)
        c = __builtin_amdgcn_wmma_f32_16x16x4_f32(
                false, a, false, b, (short)0, c, false, false);

        // Diagonal of the full matrix -> thr + 1 (before counting).
        if (ti == tj && vdiag >= 0 && vdiag < 8) c[vdiag] = maxv;

        // C layout: VGPR v holds element (row = v + 8*hi, col), col = lane&15.
        const size_t base = (size_t)(ti * TILE) * (size_t)n + (size_t)(tj * TILE) + col;
        #pragma unroll
        for (int v = 0; v < 8; ++v) {
            const int r = v + 8 * hi;
            dist[base + (size_t)r * (size_t)n] = c[v];
            // ballot bits 0-15 = row v (cols 0-15), bits 16-31 = row v+8.
            unsigned msk  = __builtin_amdgcn_ballot_w32(c[v] <= thr);
            unsigned half = hi ? (msk >> 16) : (msk & 0xFFFFu);  // branchless select
            cnt[v] += (int)__popc(half);
        }
    }

    // Flush counts once per wave: lane 0 -> rows 0..7, lane 16 -> rows 8..15.
    if (col == 0) {
        #pragma unroll
        for (int v = 0; v < 8; ++v)
            atomicAdd(&counts[ti * TILE + v + 8 * hi], cnt[v]);
    }
}

// ---------------------------------------------------------------------------
// Greedy suppression, single workgroup. Incremental count updates:
// suppressing i zeroes counts[i] and decrements counts[j] for kept j with
// dist0[i][j] <= thr (row used in place of column; the matrix is symmetric).
// Packed (count<<13)|(n-1-j) atomicMax == "max count, first index" (argmax).
// ---------------------------------------------------------------------------
__global__ __launch_bounds__(1024) void scm_greedy_kernel(
    const float* __restrict__ dist,
    const int*   __restrict__ thr_p,
    float*       __restrict__ keep,
    int*         __restrict__ counts,
    int n)
{
    const float thr = (float)(*thr_p);
    __shared__ int s_best;
    const int tid = threadIdx.x;
    const int nt  = blockDim.x;

    for (;;) {
        if (tid == 0) s_best = 0;
        __syncthreads();

        int local = 0;
        for (int j = tid; j < n; j += nt) {
            int c = counts[j];
            if (c > 0) {
                int p = (c << 13) | (n - 1 - j);
                if (p > local) local = p;
            }
        }
        if (local) atomicMax(&s_best, local);
        __syncthreads();

        const int best = s_best;
        if (best == 0) break;                    // no count > 0 -> done (uniform)
        const int i = (n - 1) - (best & 0x1FFF); // most-crowded, smallest index

        if (tid == 0) { keep[i] = 0.0f; counts[i] = 0; }
        const float* row = dist + (size_t)i * (size_t)n;
        for (int j = tid; j < n; j += nt) {
            // counts[j] > 0 implies j still kept; counts[j]==0 kept points
            // cannot have i as an in-threshold neighbor (would make count>=1).
            if (j != i && counts[j] > 0 && row[j] <= thr) counts[j] -= 1;
        }
        __syncthreads();
    }
}

// ---------------------------------------------------------------------------
extern "C" void kernel_launch(void* const* d_in, const int* in_sizes, int n_in,
                              void* d_out, int out_size, void* d_ws, size_t ws_size,
                              hipStream_t stream) {
    const float* pts   = (const float*)d_in[0];   // [n,2] float32
    const int*   thr   = (const int*)d_in[1];     // scalar int threshold
    const int    n     = in_sizes[0] / 2;         // 8192
    const int    ntile = n / TILE;                // 512

    float* dist  = (float*)d_out;                         // [n*n] dist0
    float* keep  = (float*)d_out + (size_t)n * (size_t)n; // [n] keep mask
    int*   cnts  = (int*)d_ws;                            // [n] scratch counts

    scm_init_kernel<<<(n + 255) / 256, 256, 0, stream>>>(keep, cnts, n);

    // 8 waves/block, each wave does TJ_PER_WAVE j-tiles.
    const int jslots = (ntile + TJ_PER_WAVE - 1) / TJ_PER_WAVE;   // 32
    dim3 block(32, 8);
    dim3 grid((jslots + (int)block.y - 1) / (int)block.y, ntile); // (4, 512)
    scm_dist_wmma_kernel<<<grid, block, 0, stream>>>(pts, thr, dist, cnts, n, ntile);

    scm_greedy_kernel<<<1, 1024, 0, stream>>>(dist, thr, keep, cnts, n);
}